// RNNTLoss_74431783240209
// MI455X (gfx1250) — compile-verified
//
#include <hip/hip_runtime.h>
#include <hip/hip_bf16.h>
#include <stdint.h>

// Problem constants (fixed by the reference).
#define RN_B 8
#define RN_T 200
#define RN_U 100
#define RN_U1 101
#define RN_V 512
#define ROWS (RN_B * RN_T * RN_U1)   // 161600 lattice cells
#define WAVES_PER_BLOCK 8

typedef __attribute__((ext_vector_type(2))) float v2f;
typedef __attribute__((ext_vector_type(8))) float v8f;

// ---------------------------------------------------------------------------
// Pass 1: streaming log-softmax over V=512 per (b,t,u) cell.
// One wave32 per row. Row is staged global->LDS with async b128 loads
// (ASYNCcnt path), max via shuffle tree, sum(exp) via V_WMMA_F32_16X16X4_F32.
// Writes lp_blank[b,t,u] and lp_label[b,t,u] (u<U) to workspace.
// ---------------------------------------------------------------------------
__global__ __launch_bounds__(256) void rnnt_logsoftmax_kernel(
    const float* __restrict__ logits, const int* __restrict__ y,
    float* __restrict__ lp_blank, float* __restrict__ lp_label) {
#if __HIP_DEVICE_COMPILE__
  __shared__ float smem[WAVES_PER_BLOCK * RN_V];

  const int wave = threadIdx.x >> 5;
  const int lane = threadIdx.x & 31;
  const int r = blockIdx.x * WAVES_PER_BLOCK + wave;  // grid sized exactly

  const float* rowp = logits + (size_t)r * RN_V;
  float* wptr = &smem[wave * RN_V];
  // Low 32 bits of a generic LDS address are the LDS byte offset.
  const unsigned ldsBase = (unsigned)(uintptr_t)wptr;

  // Stage 2048B row into LDS: 4 x (32 lanes x b128) async copies.
#pragma unroll
  for (int i = 0; i < 4; ++i) {
    unsigned goff = (unsigned)(i * 512 + lane * 16);
    asm volatile("global_load_async_to_lds_b128 %0, %1, %2 offset:0"
                 :
                 : "v"(ldsBase + goff), "v"(goff), "s"(rowp)
                 : "memory");
  }
  asm volatile("s_wait_asynccnt 0x0" ::: "memory");

  // Each lane pulls its 16 floats (4 x float4, 128 floats apart) from LDS.
  float x[16];
#pragma unroll
  for (int i = 0; i < 4; ++i) {
    float4 v = *reinterpret_cast<const float4*>(&wptr[i * 128 + lane * 4]);
    x[4 * i + 0] = v.x; x[4 * i + 1] = v.y;
    x[4 * i + 2] = v.z; x[4 * i + 3] = v.w;
  }

  // Row max: lane-local then wave32 xor-shuffle tree.
  float m = x[0];
#pragma unroll
  for (int i = 1; i < 16; ++i) m = fmaxf(m, x[i]);
#pragma unroll
  for (int s = 1; s < 32; s <<= 1) m = fmaxf(m, __shfl_xor(m, s));

  // Per-lane partial sum of exp(x - m).
  float partial = 0.f;
#pragma unroll
  for (int i = 0; i < 16; ++i) partial += __expf(x[i] - m);

  // Cross-lane sum via WMMA: A(16x4 f32) with A[m][0]=p[lane m],
  // A[m][2]=p[lane m+16]; B = ones(4x16). D[m][n] = p[m] + p[m+16].
  v2f a; a[0] = partial; a[1] = 0.f;
  v2f bm; bm[0] = 1.f; bm[1] = 1.f;
  v8f c = {};
  v8f d = __builtin_amdgcn_wmma_f32_16x16x4_f32(
      false, a, false, bm, (short)0, c, false, false);
  float s2 = d[0] + d[1] + d[2] + d[3] + d[4] + d[5] + d[6] + d[7];
  float S = s2 + __shfl_xor(s2, 16);  // full 32-lane sum, all lanes
  float logS = __logf(S);

  if (lane == 0) {
    const int u = r % RN_U1;
    const int bt = r / RN_U1;        // = b*T + t
    lp_blank[r] = wptr[0] - m - logS;
    if (u < RN_U) {
      const int b = bt / RN_T;
      const int yv = y[b * RN_U + u];            // in [1, V-1]
      lp_label[bt * RN_U + u] = wptr[yv] - m - logS;
    }
  }
#endif
}

// ---------------------------------------------------------------------------
// Pass 2: anti-diagonal wavefront over the (T, U+1) lattice, one block per b.
// Thread u owns column u; acol[u] holds alpha[d-u][u] after diagonal d.
// ---------------------------------------------------------------------------
__global__ __launch_bounds__(128) void rnnt_alpha_kernel(
    const float* __restrict__ lp_blank, const float* __restrict__ lp_label,
    const int* __restrict__ logit_lens, const int* __restrict__ y_lens,
    float* __restrict__ per_loss) {
#if __HIP_DEVICE_COMPILE__
  __shared__ float acol[RN_U1];

  const int b = blockIdx.x;
  const int u = threadIdx.x;
  const int tend = logit_lens[b] - 1;
  const int uend = y_lens[b];
  const float* lpb = lp_blank + (size_t)b * RN_T * RN_U1;
  const float* lpl = lp_label + (size_t)b * RN_T * RN_U;

  for (int dgn = 0; dgn < RN_T + RN_U; ++dgn) {
    const bool act = (u <= RN_U) && (u <= dgn) && (dgn - u < RN_T);
    float a_self = 0.f, a_left = 0.f;
    if (act) {
      a_self = acol[u];
      a_left = (u > 0) ? acol[u - 1] : 0.f;
    }
    __syncthreads();
    if (act) {
      const int t = dgn - u;
      float val;
      if (t == 0) {
        val = (u == 0) ? 0.f : a_left + lpl[u - 1];          // cumsum row 0
      } else if (u == 0) {
        val = a_self + lpb[(t - 1) * RN_U1];                  // blank only
      } else {
        float xA = a_self + lpb[(t - 1) * RN_U1 + u];         // from below
        float xB = a_left + lpl[t * RN_U + (u - 1)];          // from left
        float mx = fmaxf(xA, xB);
        val = mx + log1pf(__expf(-fabsf(xA - xB)));           // logaddexp
      }
      acol[u] = val;
      if (t == tend && u == uend) {
        float ll = val + lpb[tend * RN_U1 + uend];            // final blank
        per_loss[b] = -ll / (float)uend;                      // / y_lens[b]
      }
    }
    __syncthreads();
  }
#endif
}

// ---------------------------------------------------------------------------
// Pass 3: deterministic mean over B=8 per-utterance losses.
// ---------------------------------------------------------------------------
__global__ void rnnt_finalize_kernel(const float* __restrict__ per_loss,
                                     float* __restrict__ out) {
#if __HIP_DEVICE_COMPILE__
  if (threadIdx.x == 0 && blockIdx.x == 0) {
    float s = 0.f;
#pragma unroll
    for (int b = 0; b < RN_B; ++b) s += per_loss[b];
    out[0] = s / (float)RN_B;
  }
#endif
}

extern "C" void kernel_launch(void* const* d_in, const int* in_sizes, int n_in,
                              void* d_out, int out_size, void* d_ws, size_t ws_size,
                              hipStream_t stream) {
  const float* logits = (const float*)d_in[0];
  const int* y = (const int*)d_in[1];
  const int* logit_lens = (const int*)d_in[2];
  const int* y_lens = (const int*)d_in[3];

  float* ws = (float*)d_ws;
  float* lp_blank = ws;                                  // B*T*(U+1) floats
  float* lp_label = lp_blank + (size_t)RN_B * RN_T * RN_U1;  // B*T*U floats
  float* per_loss = lp_label + (size_t)RN_B * RN_T * RN_U;   // B floats

  // Pass 1: one wave per lattice row; 8 waves per 256-thread block.
  rnnt_logsoftmax_kernel<<<ROWS / WAVES_PER_BLOCK, 256, 0, stream>>>(
      logits, y, lp_blank, lp_label);

  // Pass 2: one block per batch element, 128 threads cover U+1 = 101 columns.
  rnnt_alpha_kernel<<<RN_B, 128, 0, stream>>>(lp_blank, lp_label, logit_lens,
                                              y_lens, per_loss);

  // Pass 3: scalar mean.
  rnnt_finalize_kernel<<<1, 32, 0, stream>>>(per_loss, (float*)d_out);
}